// LeakageLayer_18863496364314
// MI455X (gfx1250) — compile-verified
//
#include <hip/hip_runtime.h>
#include <hip/hip_bf16.h>
#include <math.h>

typedef __attribute__((ext_vector_type(2))) float v2f;
typedef __attribute__((ext_vector_type(8))) float v8f;

#define GMM_BATCH 4096
#define GMM_DIM   512
#define GMM_TILE  32      // rows per workgroup strip == cols per j tile
#define GMM_KC    64      // K chunk per LDS buffer (double-buffered)
#define GMM_LDSS  68      // padded stride: 68 % 64 banks = 4 -> conflict-free; 272B rows, 16B aligned

// ---------------------------------------------------------------------------
// CDNA5 async global->LDS (ASYNCcnt-tracked, bypasses VGPRs).
// LDS destination address = low 32 bits of the flat pointer (LDS aperture).
// ---------------------------------------------------------------------------
__device__ __forceinline__ void async_load_b128(unsigned lds_off, const float* g) {
  unsigned long long ga = (unsigned long long)(size_t)g;
  asm volatile("global_load_async_to_lds_b128 %0, %1, off"
               :: "v"(lds_off), "v"(ga)
               : "memory");
}
__device__ __forceinline__ void wait_async0() {
  asm volatile("s_wait_asynccnt 0x0" ::: "memory");
}

// ---------------------------------------------------------------------------
// q[i] = sum_k mu[i,k]^2   (raw, 1/s applied later)
// ---------------------------------------------------------------------------
__global__ __launch_bounds__(256) void gmm_q_kernel(const float* __restrict__ mu,
                                                    float* __restrict__ q) {
  const int wave = threadIdx.x >> 5;
  const int lane = threadIdx.x & 31;
  const int row  = blockIdx.x * 8 + wave;
  const float* r = mu + (size_t)row * GMM_DIM;
  float s = 0.0f;
  for (int c = lane; c < GMM_DIM; c += 32) {
    float v = r[c];
    s += v * v;
  }
  for (int m = 16; m >= 1; m >>= 1) s += __shfl_xor(s, m, 32);
  if (lane == 0) q[row] = s;
}

// ---------------------------------------------------------------------------
// 32-row strip per workgroup; loop over 128 column tiles. Per tile, compute
// the 32x32 block of G = mu @ mu^T with V_WMMA_F32_16X16X4_F32 (fp32 matrix
// pipe), staging K-chunks into double-buffered LDS with async global->LDS DMA
// overlapped against the WMMA K-loop. Then accumulate
// exp((G_ij - 0.5*(q_i+q_j))/s) into per-row register sums; finally
// lse[i] = log(rowsum_i) (logsumexp max == 0 via the diagonal term).
// ---------------------------------------------------------------------------
__global__ __launch_bounds__(128) void gmm_tile_kernel(const float* __restrict__ mu,
                                                       const float* __restrict__ q,
                                                       const float* __restrict__ sigma,
                                                       float* __restrict__ lse) {
  __shared__ float sA[2][GMM_TILE][GMM_LDSS];
  __shared__ float sB[2][GMM_TILE][GMM_LDSS];
  __shared__ float rowpart[4][16];

  const int t    = threadIdx.x;           // 0..127
  const int lane = t & 31;
  const int w    = t >> 5;                // wave 0..3
  const int subI = w >> 1;                // which 16-row subtile
  const int subJ = w & 1;                 // which 16-col subtile
  const int m    = lane & 15;             // row (A) / col (B,C) within subtile
  const int hi   = lane >> 4;             // half-wave
  const int hi2  = hi << 1;               // K offset for f32 16x16x4 fragments
  const int i0   = blockIdx.x * GMM_TILE;

  const float inv_s = 1.0f / sigma[0];

  // q_i for this wave's 8 C-matrix rows (invariant over the j loop)
  float qiv[8];
  {
    const int base = i0 + subI * 16 + (hi << 3);
    for (int r = 0; r < 8; ++r) qiv[r] = q[base + r];
  }

  float rsum[8];
  for (int r = 0; r < 8; ++r) rsum[r] = 0.0f;

  // staging assignment: 4 threads per row, 4 b128 transfers per matrix chunk
  const int lr = t >> 2;                  // 0..31 row
  const int lc = t & 3;                   // quad lane

  unsigned aoff[2], boff[2];
  aoff[0] = (unsigned)(size_t)&sA[0][lr][0];
  aoff[1] = (unsigned)(size_t)&sA[1][lr][0];
  boff[0] = (unsigned)(size_t)&sB[0][lr][0];
  boff[1] = (unsigned)(size_t)&sB[1][lr][0];

  const int ar = subI * 16 + m;           // A fragment row
  const int br = subJ * 16 + m;           // B fragment "column" row

  const float* aRow = &mu[(size_t)(i0 + lr) * GMM_DIM];

  for (int jt = 0; jt < GMM_BATCH / GMM_TILE; ++jt) {
    const float* bRow = &mu[(size_t)(jt * GMM_TILE + lr) * GMM_DIM];
    v8f acc = {};

    // prologue: async-DMA chunk 0 into buffer 0
    for (int qq = 0; qq < GMM_KC / 16; ++qq) {
      const int c4 = lc + qq * 4;
      async_load_b128(aoff[0] + c4 * 16, aRow + c4 * 4);
      async_load_b128(boff[0] + c4 * 16, bRow + c4 * 4);
    }

    for (int c = 0; c < GMM_DIM / GMM_KC; ++c) {
      wait_async0();        // this wave's chunk-c DMA landed in LDS
      __syncthreads();      // all waves' chunk-c landed; chunk-(c-1) reads done

      if (c + 1 < GMM_DIM / GMM_KC) {
        const int nb = (c + 1) & 1;
        const int k0 = (c + 1) * GMM_KC;
        for (int qq = 0; qq < GMM_KC / 16; ++qq) {
          const int c4 = lc + qq * 4;
          async_load_b128(aoff[nb] + c4 * 16, aRow + k0 + c4 * 4);
          async_load_b128(boff[nb] + c4 * 16, bRow + k0 + c4 * 4);
        }
      }

      // compute chunk c while chunk c+1 streams in: 16 x v_wmma_f32_16x16x4_f32
      const int cb = c & 1;
      for (int kk = 0; kk < GMM_KC; kk += 4) {
        v2f a = *reinterpret_cast<const v2f*>(&sA[cb][ar][kk + hi2]);
        v2f b = *reinterpret_cast<const v2f*>(&sB[cb][br][kk + hi2]);
        acc = __builtin_amdgcn_wmma_f32_16x16x4_f32(
            /*neg_a=*/false, a, /*neg_b=*/false, b,
            /*c_mod=*/(short)0, acc, /*reuse_a=*/false, /*reuse_b=*/false);
      }
    }

    // expo = (G - 0.5*(q_i + q_j)) / s ; accumulate exp per row
    const float qj = q[jt * GMM_TILE + subJ * 16 + m];
    for (int r = 0; r < 8; ++r) {
      const float e = (acc[r] - 0.5f * (qiv[r] + qj)) * inv_s;
      rsum[r] += expf(e);
    }
  }

  // reduce the 16 columns held by each half-wave
  for (int r = 0; r < 8; ++r) {
    float v = rsum[r];
    v += __shfl_xor(v, 1, 32);
    v += __shfl_xor(v, 2, 32);
    v += __shfl_xor(v, 4, 32);
    v += __shfl_xor(v, 8, 32);
    rsum[r] = v;
  }
  if (m == 0) {  // lane 0 (rows 0..7) and lane 16 (rows 8..15)
    for (int r = 0; r < 8; ++r) rowpart[w][hi * 8 + r] = rsum[r];
  }
  __syncthreads();
  if (t < 32) {
    const int mm    = t & 15;
    const int wbase = (t < 16) ? 0 : 2;   // waves {0,1} -> rows 0..15, {2,3} -> 16..31
    const float v   = rowpart[wbase][mm] + rowpart[wbase + 1][mm];
    lse[i0 + t] = logf(v);
  }
}

// ---------------------------------------------------------------------------
// h = D/2 + log(B) + (D/2)*log(2*pi*s) - (1/B) * sum_i lse_i ; out = {h, h}
// ---------------------------------------------------------------------------
__global__ __launch_bounds__(256) void gmm_final_kernel(const float* __restrict__ lse,
                                                        const float* __restrict__ sigma,
                                                        float* __restrict__ out) {
  __shared__ float red[256];
  float s = 0.0f;
  for (int i = threadIdx.x; i < GMM_BATCH; i += 256) s += lse[i];
  red[threadIdx.x] = s;
  __syncthreads();
  for (int step = 128; step >= 1; step >>= 1) {
    if (threadIdx.x < step) red[threadIdx.x] += red[threadIdx.x + step];
    __syncthreads();
  }
  if (threadIdx.x == 0) {
    const float sig = sigma[0];
    const float h = 0.5f * (float)GMM_DIM
                  + logf((float)GMM_BATCH)
                  + 0.5f * (float)GMM_DIM * logf(2.0f * 3.14159265358979323846f * sig)
                  - red[0] / (float)GMM_BATCH;
    out[0] = h;
    out[1] = h;
  }
}

extern "C" void kernel_launch(void* const* d_in, const int* in_sizes, int n_in,
                              void* d_out, int out_size, void* d_ws, size_t ws_size,
                              hipStream_t stream) {
  (void)in_sizes; (void)n_in; (void)out_size; (void)ws_size;
  // inputs: d_in[0]=info (unused by reference), d_in[1]=codewords, d_in[2]=sigma
  const float* mu    = (const float*)d_in[1];
  const float* sigma = (const float*)d_in[2];
  float* q   = (float*)d_ws;              // [4096]
  float* lse = q + GMM_BATCH;             // [4096]
  float* out = (float*)d_out;             // [2]

  gmm_q_kernel<<<GMM_BATCH / 8, 256, 0, stream>>>(mu, q);
  gmm_tile_kernel<<<GMM_BATCH / GMM_TILE, 128, 0, stream>>>(mu, q, sigma, lse);
  gmm_final_kernel<<<1, 256, 0, stream>>>(lse, sigma, out);
}